// DBASolver_54812372631803
// MI455X (gfx1250) — compile-verified
//
#include <hip/hip_runtime.h>
#include <hip/hip_bf16.h>

#define BB 32
#define NN 65536
#define KK 2

#define BLOCKS_Y 32
#define WPB (BLOCKS_Y * 8)     // waves per batch = 256
#define CH 16                  // n's per wave-iteration
#define ITERS (NN / CH / WPB)  // 16

// per-wave LDS buffer (floats): U[16n][2k][16] | W[32] | D[32]
#define LDS_U 1024
#define LDS_BUF (LDS_U + 32 + 32)    // 1088 floats = 4352 B

typedef __attribute__((ext_vector_type(2))) float v2f;
typedef __attribute__((ext_vector_type(4))) float v4f;
typedef __attribute__((ext_vector_type(8))) float v8f;
typedef __attribute__((ext_vector_type(2))) int   v2i;

#if __has_builtin(__builtin_amdgcn_global_load_async_to_lds_b64) && \
    __has_builtin(__builtin_amdgcn_global_load_async_to_lds_b32) && \
    __has_builtin(__builtin_amdgcn_s_wait_asynccnt)
#define USE_ASYNC_LDS 1
#else
#define USE_ASYNC_LDS 0
#endif

#define AS1 __attribute__((address_space(1)))
#define AS3 __attribute__((address_space(3)))

// ---------------------------------------------------------------------------
// Kernel 1: streaming WMMA reduction, augmented-vector LDS staging, LDS
// double-buffer fed by GLOBAL_LOAD_ASYNC_TO_LDS (ASYNCcnt) when available.
//   c1 += sum_{n,k} (w*u) (x) u,   u = [Jp(6), r, 0...]   -> H_pp | g_p
//   c2 += sum_{n}  (inv*v) (x) v,  v = [H_pd(6), g_d, 0.] -> Schur | g_corr
// U[dn][k][0..5]=Jp, [6]=r, [7..15]=0: every gather is one unconditional
// ds load (slot 6 doubles as the r / g_d element). EXEC never changes.
// WMMA A/B layout: (lane l, vgpr v) <-> (dim = l%16, K-sample = 2*(l/16)+v).
// ---------------------------------------------------------------------------
__global__ __launch_bounds__(256) void dba_reduce_k(
    const float* __restrict__ r, const float* __restrict__ w,
    const float* __restrict__ Jp, const float* __restrict__ Jd,
    const float* __restrict__ lmbda_p, float* __restrict__ partial)
{
    const int b    = blockIdx.x;
    const int lane = threadIdx.x & 31;
    const int warp = threadIdx.x >> 5;
    const int wave_id = blockIdx.y * 8 + warp;          // [0, WPB)
    const int e = lane & 15;                            // vector element (M/N)
    const int h = lane >> 4;                            // lane half -> K hi bit
    const float lm = lmbda_p[0];

    const size_t baseNK = (size_t)b * NN * KK;
    const float* rb  = r  + baseNK;
    const float* wb  = w  + baseNK;
    const float* jdb = Jd + baseNK;
    const float* jpb = Jp + baseNK * 6;

    __shared__ float lds[8 * 2 * LDS_BUF];
    float* Sbuf0 = lds + warp * (2 * LDS_BUF);
    float* Sbuf1 = Sbuf0 + LDS_BUF;

    // ---- zero-fill augmented pad slots 7..15 in both buffers
    {
        float* Uz0 = Sbuf0 + lane * 16;
        float* Uz1 = Sbuf1 + lane * 16;
        Uz0[7] = 0.f;
        *(v4f*)(Uz0 + 8)  = (v4f){0.f, 0.f, 0.f, 0.f};
        *(v4f*)(Uz0 + 12) = (v4f){0.f, 0.f, 0.f, 0.f};
        Uz1[7] = 0.f;
        *(v4f*)(Uz1 + 8)  = (v4f){0.f, 0.f, 0.f, 0.f};
        *(v4f*)(Uz1 + 12) = (v4f){0.f, 0.f, 0.f, 0.f};
    }

    // ---- loop-invariant staging scatter offsets: flat jp idx f -> (f/6)*16+f%6
    const int f0 = 2 * lane;
    const int f1 = 2 * lane + 64;
    const int f2 = 2 * lane + 128;
    const int u0 = (f0 / 6) * 16 + (f0 % 6);
    const int u1 = (f1 / 6) * 16 + (f1 % 6);
    const int u2 = (f2 / 6) * 16 + (f2 % 6);

    v8f c1 = {0.f,0.f,0.f,0.f,0.f,0.f,0.f,0.f};
    v8f c2 = {0.f,0.f,0.f,0.f,0.f,0.f,0.f,0.f};

#if USE_ASYNC_LDS
    // ---- async stage of chunk `cit` into buffer `Sb` (6 ops on ASYNCcnt)
    #define STAGE_ASYNC(cit, Sb)                                               \
        do {                                                                   \
            const int n0_ = (wave_id + (cit) * WPB) * CH;                      \
            const v2f* jp2_ = (const v2f*)(jpb + (size_t)n0_ * 12);            \
            __builtin_amdgcn_global_load_async_to_lds_b64(                     \
                (AS1 v2i*)(jp2_ + lane),      (AS3 v2i*)((Sb) + u0), 0, 0);    \
            __builtin_amdgcn_global_load_async_to_lds_b64(                     \
                (AS1 v2i*)(jp2_ + lane + 32), (AS3 v2i*)((Sb) + u1), 0, 0);    \
            __builtin_amdgcn_global_load_async_to_lds_b64(                     \
                (AS1 v2i*)(jp2_ + lane + 64), (AS3 v2i*)((Sb) + u2), 0, 0);    \
            __builtin_amdgcn_global_load_async_to_lds_b32(                     \
                (AS1 int*)(wb + (size_t)n0_ * 2 + lane),                       \
                (AS3 int*)((Sb) + LDS_U + lane), 0, 0);                        \
            __builtin_amdgcn_global_load_async_to_lds_b32(                     \
                (AS1 int*)(rb + (size_t)n0_ * 2 + lane),                       \
                (AS3 int*)((Sb) + lane * 16 + 6), 0, 0);                       \
            __builtin_amdgcn_global_load_async_to_lds_b32(                     \
                (AS1 int*)(jdb + (size_t)n0_ * 2 + lane),                      \
                (AS3 int*)((Sb) + LDS_U + 32 + lane), 0, 0);                   \
        } while (0)

    STAGE_ASYNC(0, Sbuf0);
#else
    // ---- fallback: register staging (coalesced loads, ds stores)
    v2f jA, jB, jC;
    float wA, rA, dA;
    {
        const int n0 = wave_id * CH;
        const v2f* jp2 = (const v2f*)(jpb + (size_t)n0 * 12);
        jA = jp2[lane];
        jB = jp2[lane + 32];
        jC = jp2[lane + 64];
        wA = wb[(size_t)n0 * 2 + lane];
        rA = rb[(size_t)n0 * 2 + lane];
        dA = jdb[(size_t)n0 * 2 + lane];
    }
#endif

#pragma clang loop unroll(disable)
    for (int it = 0; it < ITERS; ++it) {
        float* cur = (it & 1) ? Sbuf1 : Sbuf0;

#if USE_ASYNC_LDS
        {
            float* nxt = (it & 1) ? Sbuf0 : Sbuf1;
            const int itn = (it + 1 < ITERS) ? (it + 1) : (ITERS - 1);
            STAGE_ASYNC(itn, nxt);
            const int itp = (it + 2 < ITERS) ? (it + 2) : (ITERS - 1);
            const int n0p = (wave_id + itp * WPB) * CH;
            __builtin_prefetch(jpb + (size_t)n0p * 12 + lane * 6, 0, 1);
            // async loads complete in order: wait until only the 6 just-issued
            // remain -> chunk `it` fully in `cur`.
            __builtin_amdgcn_s_wait_asynccnt(6);
        }
#else
        {
            // stage current chunk regs -> LDS (same-wave LDS is in-order)
            float* U = cur;
            *(v2f*)(U + u0) = jA;
            *(v2f*)(U + u1) = jB;
            *(v2f*)(U + u2) = jC;
            U[lane * 16 + 6] = rA;
            U[LDS_U + lane] = wA;
            U[LDS_U + 32 + lane] = dA;
            const int itn = (it + 1 < ITERS) ? (it + 1) : (ITERS - 1);
            const int n0n = (wave_id + itn * WPB) * CH;
            const v2f* jp2 = (const v2f*)(jpb + (size_t)n0n * 12);
            jA = jp2[lane];
            jB = jp2[lane + 32];
            jC = jp2[lane + 64];
            wA = wb[(size_t)n0n * 2 + lane];
            rA = rb[(size_t)n0n * 2 + lane];
            dA = jdb[(size_t)n0n * 2 + lane];
            const int itp = (it + 2 < ITERS) ? (it + 2) : (ITERS - 1);
            const int n0p = (wave_id + itp * WPB) * CH;
            __builtin_prefetch(jpb + (size_t)n0p * 12 + lane * 6, 0, 1);
        }
#endif

        const float* U = cur;
        const float* W = cur + LDS_U;
        const float* D = cur + LDS_U + 32;

        // ---- S1: 8 WMMAs; WMMA q: lane sample (dn = 2q + h, k = v)
        const float* Ub1 = U + h * 32 + e;   // + q*64 (+ v*16)
        const float* Wb1 = W + h * 2;        // + q*4  (+ v)
#pragma unroll
        for (int q = 0; q < 8; ++q) {
            const float ua  = Ub1[q * 64];
            const float ubv = Ub1[q * 64 + 16];
            const v2f   wp  = *(const v2f*)(Wb1 + q * 4);
            v2f bv, av;
            bv[0] = ua;        bv[1] = ubv;
            av[0] = wp.x * ua; av[1] = wp.y * ubv;
            c1 = __builtin_amdgcn_wmma_f32_16x16x4_f32(
                     false, av, false, bv, (short)0, c1, false, false);
        }

        // ---- S2: 4 WMMAs; WMMA t: lane sample dn = 4t + 2h + v
        const float* Ub2 = U + h * 64 + e;   // + (4t+v)*32 (+ k*16)
        const float* Wb2 = W + h * 4;        // + (4t+v)*2
        const float* Db2 = D + h * 4;
#pragma unroll
        for (int t = 0; t < 4; ++t) {
            v2f bv, av;
#pragma unroll
            for (int v = 0; v < 2; ++v) {
                const int o32 = (4 * t + v) * 32;
                const int o2  = (4 * t + v) * 2;
                const float up0 = Ub2[o32];
                const float up1 = Ub2[o32 + 16];
                const v2f   wp  = *(const v2f*)(Wb2 + o2);
                const v2f   dp  = *(const v2f*)(Db2 + o2);
                const float a0 = wp.x * dp.x, a1 = wp.y * dp.y;
                const float hdd = a0 * dp.x + a1 * dp.y + lm;
                const float inv = __builtin_amdgcn_rcpf(hdd + 1e-8f);
                const float val = a0 * up0 + a1 * up1;  // hpd[e] | g_d | 0
                bv[v] = val;
                av[v] = inv * val;
            }
            c2 = __builtin_amdgcn_wmma_f32_16x16x4_f32(
                     false, av, false, bv, (short)0, c2, false, false);
        }
    }

    // C layout: VGPR i, lanes 0..15 hold row M=i, col N=lane. Rows 0..5, cols 0..6.
    if (lane < 7) {
        float* pp = partial + ((size_t)b * WPB + wave_id) * 84;
#pragma unroll
        for (int i = 0; i < 6; ++i) {
            pp[i * 7 + lane]      = c1[i];
            pp[42 + i * 7 + lane] = c2[i];
        }
    }
}

// ---------------------------------------------------------------------------
// Kernel 2: deterministic fixed-order reduction of per-wave partials.
// ---------------------------------------------------------------------------
__global__ void dba_sum_k(const float* __restrict__ partial,
                          float* __restrict__ acc)
{
    const int idx = blockIdx.x * blockDim.x + threadIdx.x;
    if (idx >= BB * 84) return;
    const int b = idx / 84;
    const int q = idx % 84;
    float s = 0.f;
    for (int wv = 0; wv < WPB; ++wv)
        s += partial[((size_t)b * WPB + wv) * 84 + q];
    acc[idx] = s;
}

// ---------------------------------------------------------------------------
// Kernel 3: per-batch 6x6 solve with partial pivoting. 32 lanes = 32 batches.
// ---------------------------------------------------------------------------
__global__ void dba_solve_k(const float* __restrict__ acc,
                            const float* __restrict__ lmbda_p,
                            float* __restrict__ pose_out)
{
    const int b = threadIdx.x;
    if (b >= BB) return;
    const float lm = lmbda_p[0];
    const float* a1 = acc + (size_t)b * 84;
    const float* a2 = a1 + 42;

    float H[6][7];
    for (int i = 0; i < 6; ++i) {
        for (int j = 0; j < 6; ++j)
            H[i][j] = a1[i * 7 + j] - a2[i * 7 + j] +
                      ((i == j) ? (lm + 1e-4f) : 0.f);
        H[i][6] = a1[i * 7 + 6] - a2[i * 7 + 6];
    }
    for (int col = 0; col < 6; ++col) {
        int piv = col;
        float best = fabsf(H[col][col]);
        for (int rr = col + 1; rr < 6; ++rr) {
            float v = fabsf(H[rr][col]);
            if (v > best) { best = v; piv = rr; }
        }
        if (piv != col) {
            for (int j = col; j < 7; ++j) {
                float t = H[col][j]; H[col][j] = H[piv][j]; H[piv][j] = t;
            }
        }
        const float invp = 1.0f / H[col][col];
        for (int rr = col + 1; rr < 6; ++rr) {
            const float f = H[rr][col] * invp;
            for (int j = col; j < 7; ++j) H[rr][j] -= f * H[col][j];
        }
    }
    float x[6];
    for (int i = 5; i >= 0; --i) {
        float s = H[i][6];
        for (int j = i + 1; j < 6; ++j) s -= H[i][j] * x[j];
        x[i] = s / H[i][i];
    }
    for (int i = 0; i < 6; ++i) pose_out[b * 6 + i] = x[i];
}

// ---------------------------------------------------------------------------
// Kernel 4: depth back-substitution, vectorized loads (inputs L2-resident).
// delta_depth = invHdd * (g_d - H_pd . delta_pose)
// ---------------------------------------------------------------------------
__global__ __launch_bounds__(256) void dba_depth_k(
    const float* __restrict__ r, const float* __restrict__ w,
    const float* __restrict__ Jp, const float* __restrict__ Jd,
    const float* __restrict__ lmbda_p, const float* __restrict__ pose,
    float* __restrict__ depth)
{
    const int b = blockIdx.y;
    const int n = blockIdx.x * blockDim.x + threadIdx.x;
    const float lm = lmbda_p[0];

    const size_t nk0 = ((size_t)b * NN + n) * KK;
    const v2f wv = *(const v2f*)(w  + nk0);
    const v2f dv = *(const v2f*)(Jd + nk0);
    const v2f rv = *(const v2f*)(r  + nk0);
    const v4f jp0 = *(const v4f*)(Jp + nk0 * 6);      // Jp[k0][0..3]
    const v4f jp1 = *(const v4f*)(Jp + nk0 * 6 + 4);  // Jp[k0][4..5], Jp[k1][0..1]
    const v4f jp2 = *(const v4f*)(Jp + nk0 * 6 + 8);  // Jp[k1][2..5]

    const float a0 = wv.x * dv.x, a1 = wv.y * dv.y;
    const float gd  = a0 * rv.x + a1 * rv.y;
    const float hdd = a0 * dv.x + a1 * dv.y + lm;
    const float inv = __builtin_amdgcn_rcpf(hdd + 1e-8f);

    const float p0 = pose[b * 6 + 0], p1 = pose[b * 6 + 1], p2 = pose[b * 6 + 2];
    const float p3 = pose[b * 6 + 3], p4 = pose[b * 6 + 4], p5 = pose[b * 6 + 5];

    float vsum = (a0 * jp0.x + a1 * jp1.z) * p0 +
                 (a0 * jp0.y + a1 * jp1.w) * p1 +
                 (a0 * jp0.z + a1 * jp2.x) * p2 +
                 (a0 * jp0.w + a1 * jp2.y) * p3 +
                 (a0 * jp1.x + a1 * jp2.z) * p4 +
                 (a0 * jp1.y + a1 * jp2.w) * p5;

    depth[(size_t)b * NN + n] = inv * (gd - vsum);
}

// ---------------------------------------------------------------------------
extern "C" void kernel_launch(void* const* d_in, const int* in_sizes, int n_in,
                              void* d_out, int out_size, void* d_ws, size_t ws_size,
                              hipStream_t stream)
{
    (void)in_sizes; (void)n_in; (void)out_size; (void)ws_size;
    const float* r  = (const float*)d_in[0];
    const float* w  = (const float*)d_in[1];
    const float* Jp = (const float*)d_in[2];
    const float* Jd = (const float*)d_in[3];
    const float* lm = (const float*)d_in[4];
    float* out = (float*)d_out;                 // [B*6] pose, then [B*N] depth

    float* partial = (float*)d_ws;              // [B][WPB][84]
    float* acc     = partial + (size_t)BB * WPB * 84;

    dim3 g1(BB, BLOCKS_Y);
    dba_reduce_k<<<g1, 256, 0, stream>>>(r, w, Jp, Jd, lm, partial);

    dba_sum_k<<<(BB * 84 + 255) / 256, 256, 0, stream>>>(partial, acc);

    dba_solve_k<<<1, 32, 0, stream>>>(acc, lm, out);

    dim3 g4(NN / 256, BB);
    dba_depth_k<<<g4, 256, 0, stream>>>(r, w, Jp, Jd, lm, out, out + BB * 6);
}